// VanillaRIN_75892072121086
// MI455X (gfx1250) — compile-verified
//
#include <hip/hip_runtime.h>
#include <math.h>

#define PHI_F 1.6180339887498949f
#define Vv 32000
#define Dd 256
#define Ll 2
#define Nn 512
#define Bb 8
#define Ss 512
#define ROWS (Bb * Ss) /* 4096 */
#define LS 548         /* LDS row stride (floats): 548%64=36 -> conflict-free for 16 rows */
#define ADOFF 274      /* second-half offset within an LDS row (even -> 8B aligned pairs) */

typedef __attribute__((ext_vector_type(2))) float v2f;
typedef __attribute__((ext_vector_type(8))) float v8f;

static __device__ __forceinline__ v8f wmma_f32(v2f a, v2f b, v8f c) {
  // V_WMMA_F32_16X16X4_F32 : D = A(16x4) x B(4x16) + C(16x16)
  return __builtin_amdgcn_wmma_f32_16x16x4_f32(false, a, false, b, (short)0, c,
                                               false, false);
}

// ---------------------------------------------------------------------------
// Kernel 1: elementwise recurrence over (b,d); emits Xr/Xi for every t and the
// final carry (hr,hi) into the output tail. 2048 threads total.
// ---------------------------------------------------------------------------
__global__ void k_rec(const int* __restrict__ ids, const float* __restrict__ emb,
                      float* __restrict__ Xr, float* __restrict__ Xi,
                      float* __restrict__ hr_out, float* __restrict__ hi_out) {
  int gid = blockIdx.x * blockDim.x + threadIdx.x; // 0..2047
  int b = gid >> 8;
  int d = gid & 255;
  float hr = 0.0f, hi = 0.0f;
  for (int t = 0; t < Ss; ++t) {
    int id = ids[b * Ss + t];
    float w  = emb[(size_t)id * (2 * Dd) + d];
    float bt = emb[(size_t)id * (2 * Dd) + Dd + d];
    float wl = 1.0f + fabsf(w);
    float tphi = (float)t * PHI_F;
    float thr = hr / wl + bt + tphi;
    float thi = hi / wl + bt + tphi;
    float sr = sinf(thr), cr = cosf(thr);
    float si = sinf(thi), ci = cosf(thi);
    hr = cr * ci - sr * si;
    hi = cr * si + sr * ci;
    Xr[(size_t)(t * Bb + b) * Dd + d] = hr;
    Xi[(size_t)(t * Bb + b) * Dd + d] = hi;
  }
  hr_out[b * Dd + d] = hr;
  hi_out[b * Dd + d] = hi;
}

// ---------------------------------------------------------------------------
// Kernel 2: one fused layer. Block = 16 rows x 256 threads (8 waves).
//   phase0: stage (xr+xi) and (xr-xi) tiles in LDS
//   phase1: in-GEMM  u = (xr+xi)@inWr + (xr-xi)@inWi   (wave -> 16x64 tile)
//   phase2: ang = (u + br+bi)/wl + 2*lb + 2*t*phi  -> LDS
//   phase3: out-GEMM dr = cos@outWr - sin@outWi ; di = cos@outWi + sin@outWr
//   phase4: residual update of Xr/Xi in place (wave -> 16x32 tile)
// ---------------------------------------------------------------------------
__global__ void k_layer(float* __restrict__ Xr, float* __restrict__ Xi,
                        const float* __restrict__ lw, const float* __restrict__ lb,
                        const float* __restrict__ inWr, const float* __restrict__ inWi,
                        const float* __restrict__ inbr, const float* __restrict__ inbi,
                        const float* __restrict__ outWr, const float* __restrict__ outWi,
                        const float* __restrict__ outbr, const float* __restrict__ outbi) {
  __shared__ float lds[16 * LS];
  const int tid = threadIdx.x;
  const int r0 = blockIdx.x * 16;
  const int wave = tid >> 5;
  const int lane = tid & 31;
  const int m = lane & 15;
  const int hl = lane >> 4;

  // phase 0
  for (int idx = tid; idx < 16 * Dd; idx += 256) {
    int mm = idx >> 8;
    int k = idx & 255;
    float xr = Xr[(size_t)(r0 + mm) * Dd + k];
    float xi = Xi[(size_t)(r0 + mm) * Dd + k];
    lds[mm * LS + k] = xr + xi;
    lds[mm * LS + ADOFF + k] = xr - xi;
  }
  __syncthreads();

  // phase 1: in-GEMM, wave owns N columns [wave*64, wave*64+64)
  const int n0w = wave * 64;
  v8f cu[4];
#pragma unroll
  for (int j = 0; j < 4; ++j) cu[j] = (v8f){0, 0, 0, 0, 0, 0, 0, 0};
#pragma unroll 2
  for (int k0 = 0; k0 < Dd; k0 += 4) {
    int kA = k0 + 2 * hl;
    v2f aS = *(const v2f*)&lds[m * LS + kA];
    v2f aD = *(const v2f*)&lds[m * LS + ADOFF + kA];
#pragma unroll
    for (int j = 0; j < 4; ++j) {
      int col = n0w + j * 16 + m;
      v2f bR = {inWr[(size_t)kA * Nn + col], inWr[(size_t)(kA + 1) * Nn + col]};
      v2f bI = {inWi[(size_t)kA * Nn + col], inWi[(size_t)(kA + 1) * Nn + col]};
      cu[j] = wmma_f32(aS, bR, cu[j]);
      cu[j] = wmma_f32(aD, bI, cu[j]);
    }
  }
  __syncthreads(); // everyone done reading sum/diff tiles

  // phase 2: angle into LDS
#pragma unroll
  for (int j = 0; j < 4; ++j) {
    int n = n0w + j * 16 + m;
    float wl = 1.0f + fabsf(lw[n]);
    float lbn = lb[n];
    float bsum = inbr[n] + inbi[n];
#pragma unroll
    for (int i = 0; i < 8; ++i) {
      int rl = i + 8 * hl;
      float t = (float)((r0 + rl) >> 3); // row = t*B + b, B = 8
      float ang = (cu[j][i] + bsum) / wl + 2.0f * lbn + 2.0f * t * PHI_F;
      lds[rl * LS + n] = ang;
    }
  }
  __syncthreads();

  // phase 3: out-GEMM, wave owns D columns [wave*32, wave*32+32)
  const int n0o = wave * 32;
  v8f cdr[2], cdi[2];
#pragma unroll
  for (int s = 0; s < 2; ++s) {
    cdr[s] = (v8f){0, 0, 0, 0, 0, 0, 0, 0};
    cdi[s] = (v8f){0, 0, 0, 0, 0, 0, 0, 0};
  }
#pragma unroll 2
  for (int k0 = 0; k0 < Nn; k0 += 4) {
    int kA = k0 + 2 * hl;
    v2f a = *(const v2f*)&lds[m * LS + kA];
    v2f ac = {cosf(a.x), cosf(a.y)};
    v2f as = {sinf(a.x), sinf(a.y)};
    v2f an = {-as.x, -as.y}; // f32 WMMA has no A-neg modifier: negate manually
#pragma unroll
    for (int s = 0; s < 2; ++s) {
      int col = n0o + s * 16 + m;
      v2f bR = {outWr[(size_t)kA * Dd + col], outWr[(size_t)(kA + 1) * Dd + col]};
      v2f bI = {outWi[(size_t)kA * Dd + col], outWi[(size_t)(kA + 1) * Dd + col]};
      cdr[s] = wmma_f32(ac, bR, cdr[s]);
      cdr[s] = wmma_f32(an, bI, cdr[s]);
      cdi[s] = wmma_f32(ac, bI, cdi[s]);
      cdi[s] = wmma_f32(as, bR, cdi[s]);
    }
  }

  // phase 4: residual update
#pragma unroll
  for (int s = 0; s < 2; ++s) {
    int col = n0o + s * 16 + m;
    float obr = outbr[col], obi = outbi[col];
#pragma unroll
    for (int i = 0; i < 8; ++i) {
      int row = r0 + i + 8 * hl;
      Xr[(size_t)row * Dd + col] += cdr[s][i] + obr;
      Xi[(size_t)row * Dd + col] += cdi[s][i] + obi;
    }
  }
}

// ---------------------------------------------------------------------------
// Kernel 3: in-place (Xr,Xi) -> (Xr+Xi, Xr-Xi)
// ---------------------------------------------------------------------------
__global__ void k_pq(float* __restrict__ Xr, float* __restrict__ Xi) {
  size_t i = (size_t)blockIdx.x * blockDim.x + threadIdx.x;
  float xr = Xr[i], xi = Xi[i];
  Xr[i] = xr + xi;
  Xi[i] = xr - xi;
}

// ---------------------------------------------------------------------------
// Kernel 4: projection. logits = P@proj_Wr + Q@proj_Wi, P=xr+xi, Q=xr-xi.
// Grid (125, 256): block = 16 rows x 256 cols; wave owns a 16x32 tile.
// ---------------------------------------------------------------------------
__global__ void k_proj(const float* __restrict__ P, const float* __restrict__ Q,
                       const float* __restrict__ Wr, const float* __restrict__ Wi,
                       float* __restrict__ out) {
  __shared__ float lds[16 * LS];
  const int tid = threadIdx.x;
  const int n0 = blockIdx.x * 256;
  const int r0 = blockIdx.y * 16;
  const int wave = tid >> 5;
  const int lane = tid & 31;
  const int m = lane & 15;
  const int hl = lane >> 4;

  for (int idx = tid; idx < 16 * Dd; idx += 256) {
    int mm = idx >> 8;
    int k = idx & 255;
    lds[mm * LS + k] = P[(size_t)(r0 + mm) * Dd + k];
    lds[mm * LS + ADOFF + k] = Q[(size_t)(r0 + mm) * Dd + k];
  }
  __syncthreads();

  const int nw = n0 + wave * 32;
  v8f c0 = (v8f){0, 0, 0, 0, 0, 0, 0, 0};
  v8f c1 = (v8f){0, 0, 0, 0, 0, 0, 0, 0};
#pragma unroll 2
  for (int k0 = 0; k0 < Dd; k0 += 4) {
    int kA = k0 + 2 * hl;
    v2f aP = *(const v2f*)&lds[m * LS + kA];
    v2f aQ = *(const v2f*)&lds[m * LS + ADOFF + kA];
    const float* wr0 = Wr + (size_t)kA * Vv;
    const float* wi0 = Wi + (size_t)kA * Vv;
    v2f bR0 = {wr0[nw + m], wr0[Vv + nw + m]};
    v2f bI0 = {wi0[nw + m], wi0[Vv + nw + m]};
    v2f bR1 = {wr0[nw + 16 + m], wr0[Vv + nw + 16 + m]};
    v2f bI1 = {wi0[nw + 16 + m], wi0[Vv + nw + 16 + m]};
    c0 = wmma_f32(aP, bR0, c0);
    c0 = wmma_f32(aQ, bI0, c0);
    c1 = wmma_f32(aP, bR1, c1);
    c1 = wmma_f32(aQ, bI1, c1);
  }

#pragma unroll
  for (int i = 0; i < 8; ++i) {
    int row = r0 + i + 8 * hl; // row = t*B + b
    int t = row >> 3;
    int b = row & 7;
    size_t base = ((size_t)b * Ss + t) * (size_t)Vv; // output is (B, S, V)
    out[base + nw + m] = c0[i];
    out[base + nw + 16 + m] = c1[i];
  }
}

extern "C" void kernel_launch(void* const* d_in, const int* in_sizes, int n_in,
                              void* d_out, int out_size, void* d_ws, size_t ws_size,
                              hipStream_t stream) {
  const int* ids    = (const int*)d_in[0];
  const float* emb  = (const float*)d_in[1];
  const float* lw   = (const float*)d_in[2];
  const float* lb   = (const float*)d_in[3];
  const float* inWr = (const float*)d_in[4];
  const float* inWi = (const float*)d_in[5];
  const float* inbr = (const float*)d_in[6];
  const float* inbi = (const float*)d_in[7];
  const float* outWr = (const float*)d_in[8];
  const float* outWi = (const float*)d_in[9];
  const float* outbr = (const float*)d_in[10];
  const float* outbi = (const float*)d_in[11];
  const float* pWr  = (const float*)d_in[12];
  const float* pWi  = (const float*)d_in[13];

  float* out = (float*)d_out;
  float* hr_out = out + (size_t)Bb * Ss * Vv;
  float* hi_out = hr_out + (size_t)Bb * Dd;

  float* Xr = (float*)d_ws;                 // ROWS x Dd
  float* Xi = Xr + (size_t)ROWS * Dd;       // ROWS x Dd  (8 MB total)

  k_rec<<<8, 256, 0, stream>>>(ids, emb, Xr, Xi, hr_out, hi_out);

  for (int l = 0; l < Ll; ++l) {
    k_layer<<<ROWS / 16, 256, 0, stream>>>(
        Xr, Xi, lw + (size_t)l * Nn, lb + (size_t)l * Nn,
        inWr + (size_t)l * Dd * Nn, inWi + (size_t)l * Dd * Nn,
        inbr + (size_t)l * Nn, inbi + (size_t)l * Nn,
        outWr + (size_t)l * Nn * Dd, outWi + (size_t)l * Nn * Dd,
        outbr + (size_t)l * Dd, outbi + (size_t)l * Dd);
  }

  k_pq<<<(ROWS * Dd) / 256, 256, 0, stream>>>(Xr, Xi);

  k_proj<<<dim3(Vv / 256, ROWS / 16), 256, 0, stream>>>(Xr, Xi, pWr, pWi, out);
}